// GCN_20779051778665
// MI455X (gfx1250) — compile-verified
//
#include <hip/hip_runtime.h>

// ---------------------------------------------------------------------------
// 3-layer GCN (DGL GraphConv, norm='right') for MI455X / gfx1250.
//  - scatter-add aggregation via global_atomic_add_f32 (memory-bound part)
//  - per-layer dense GEMM via V_WMMA_F32_16X16X4_F32 (full fp32, matches ref)
// ---------------------------------------------------------------------------

#define N_NODES 50000
#define N_EDGES 600000

typedef __attribute__((ext_vector_type(2))) float v2f;
typedef __attribute__((ext_vector_type(8))) float v8f;

__global__ void zero_f32(float* __restrict__ p, int n) {
  int i = blockIdx.x * blockDim.x + threadIdx.x;
  if (i < n) p[i] = 0.0f;
}

__global__ void degree_kernel(const int* __restrict__ dst, float* __restrict__ deg) {
  int e = blockIdx.x * blockDim.x + threadIdx.x;
  if (e < N_EDGES) atomicAdd(&deg[dst[e]], 1.0f);
}

__global__ void invdeg_kernel(float* __restrict__ deg) {
  int i = blockIdx.x * blockDim.x + threadIdx.x;
  if (i < N_NODES) {
    float d = deg[i];
    deg[i] = 1.0f / fmaxf(d, 1.0f);
  }
}

// One wave32 per edge; each lane moves one float4 (lane*16B) of the 512B row.
// Coalesced 512B gather per edge, 4 f32 atomics per lane into agg[dst].
__global__ void scatter128(const float* __restrict__ h,
                           const int* __restrict__ src,
                           const int* __restrict__ dst,
                           float* __restrict__ agg) {
  unsigned gtid = blockIdx.x * blockDim.x + threadIdx.x;
  unsigned e    = gtid >> 5;
  unsigned lane = gtid & 31u;
  if (e >= N_EDGES) return;
  int s = src[e];
  int d = dst[e];
  const float4 v = ((const float4*)(h + (size_t)s * 128))[lane];
  float* a = agg + (size_t)d * 128 + lane * 4;
  atomicAdd(a + 0, v.x);
  atomicAdd(a + 1, v.y);
  atomicAdd(a + 2, v.z);
  atomicAdd(a + 3, v.w);
}

// Fused (inv_deg * agg) @ W + bias (+ReLU) using fp32 WMMA.
// Block = one 16-row node tile, DN/16 waves (one per 16-col tile of W).
// A tile staged in LDS (stride 130 floats -> conflict-free ds_load_b64:
// row stride mod 64 banks = 2, so the 16 lanes hit 16 distinct bank pairs).
template <int DK, int DN, bool RELU>
__global__ __launch_bounds__(32 * (DN / 16)) void gcn_gemm(
    const float* __restrict__ agg,     // [N_NODES x DK]
    const float* __restrict__ invdeg,  // [N_NODES]
    const float* __restrict__ W,       // [DK x DN]
    const float* __restrict__ bias,    // [DN]
    float* __restrict__ out)           // [N_NODES x DN]
{
  constexpr int NW   = DN / 16;   // waves per block
  constexpr int LSTR = DK + 2;    // padded LDS row stride (floats)
  __shared__ float As[16 * LSTR];

  const int row0 = blockIdx.x * 16;
  const int tid  = threadIdx.x;

  // Cooperative A-tile load with fused inv_deg scaling (16*DK floats).
  for (int i = tid; i < 16 * DK; i += 32 * NW) {
    int r = i / DK;
    int c = i - r * DK;
    As[r * LSTR + c] = agg[(size_t)(row0 + r) * DK + c] * invdeg[row0 + r];
  }
  __syncthreads();

  const int wv   = tid >> 5;      // column tile index
  const int lane = tid & 31;
  const int half = lane >> 4;     // 0 or 1
  const int mn   = lane & 15;     // A row (M) / B,C column (N) within tile
  const int col0 = wv * 16;

  v8f c = {};
  // ISA layouts (7.12.2):
  //   A 16x4 f32 : lane (half,mn) holds {A[mn][k+2*half], A[mn][k+2*half+1]}
  //   B  4x16 f32: VGPR v, lane -> B[v + 2*half][mn]
  //   C 16x16 f32: VGPR v, lane -> C[v + 8*half][mn]
  for (int k = 0; k < DK; k += 4) {
    const int kk = k + 2 * half;
    v2f a = *(const v2f*)&As[mn * LSTR + kk];          // 8B-aligned: LSTR,kk even
    v2f b;
    b.x = W[(size_t)kk * DN + col0 + mn];
    b.y = W[(size_t)(kk + 1) * DN + col0 + mn];
    c = __builtin_amdgcn_wmma_f32_16x16x4_f32(
        /*neg_a=*/false, a, /*neg_b=*/false, b,
        /*c_mod=*/(short)0, c, /*reuse_a=*/false, /*reuse_b=*/false);
  }

  const float bv = bias[col0 + mn];
#pragma unroll
  for (int v = 0; v < 8; ++v) {
    const int r = row0 + v + 8 * half;
    float x = c[v] + bv;
    if (RELU) x = fmaxf(x, 0.0f);
    out[(size_t)r * DN + col0 + mn] = x;
  }
}

extern "C" void kernel_launch(void* const* d_in, const int* in_sizes, int n_in,
                              void* d_out, int out_size, void* d_ws, size_t ws_size,
                              hipStream_t stream) {
  const float* feat = (const float*)d_in[0];
  const int*   src  = (const int*)d_in[1];
  const int*   dst  = (const int*)d_in[2];
  const float* W1   = (const float*)d_in[3];
  const float* b1   = (const float*)d_in[4];
  const float* W2   = (const float*)d_in[5];
  const float* b2   = (const float*)d_in[6];
  const float* W3   = (const float*)d_in[7];
  const float* b3   = (const float*)d_in[8];
  float* out = (float*)d_out;

  // Workspace: inv_deg | agg buffer | h ping buffer  (~51.4 MB f32)
  float* invdeg = (float*)d_ws;
  float* bufA   = invdeg + N_NODES;                 // aggregation target
  float* bufB   = bufA + (size_t)N_NODES * 128;     // hidden activations

  const int TB = 256;
  const int NF = N_NODES * 128;
  const int scat_blocks = (N_EDGES * 32) / TB;      // 75000 exactly
  const int gemm_blocks = N_NODES / 16;             // 3125 exactly

  // in-degree -> inv_deg (in place)
  zero_f32<<<(N_NODES + TB - 1) / TB, TB, 0, stream>>>(invdeg, N_NODES);
  degree_kernel<<<(N_EDGES + TB - 1) / TB, TB, 0, stream>>>(dst, invdeg);
  invdeg_kernel<<<(N_NODES + TB - 1) / TB, TB, 0, stream>>>(invdeg);

  // layer 1: feat -> bufB
  zero_f32<<<(NF + TB - 1) / TB, TB, 0, stream>>>(bufA, NF);
  scatter128<<<scat_blocks, TB, 0, stream>>>(feat, src, dst, bufA);
  gcn_gemm<128, 128, true><<<gemm_blocks, 256, 0, stream>>>(bufA, invdeg, W1, b1, bufB);

  // layer 2: bufB -> bufB (agg consumed before overwrite; same stream orders it)
  zero_f32<<<(NF + TB - 1) / TB, TB, 0, stream>>>(bufA, NF);
  scatter128<<<scat_blocks, TB, 0, stream>>>(bufB, src, dst, bufA);
  gcn_gemm<128, 128, true><<<gemm_blocks, 256, 0, stream>>>(bufA, invdeg, W2, b2, bufB);

  // layer 3: bufB -> d_out [50000 x 64]
  zero_f32<<<(NF + TB - 1) / TB, TB, 0, stream>>>(bufA, NF);
  scatter128<<<scat_blocks, TB, 0, stream>>>(bufB, src, dst, bufA);
  gcn_gemm<128, 64, false><<<gemm_blocks, 128, 0, stream>>>(bufA, invdeg, W3, b3, out);
}